// GCN_73942156968188
// MI455X (gfx1250) — compile-verified
//
#include <hip/hip_runtime.h>

// ---------------------------------------------------------------------------
// GCN pipeline for MI455X (gfx1250, wave32).
// - All matmuls use V_WMMA_F32_16X16X4_F32 (fp32, workload is memory-bound:
//   ~330MB mandatory traffic vs <8 GFLOP, so fp32 WMMA is the right path).
// - B operands (<=20KB per batch) are DMA'd to LDS by the Tensor Data Mover
//   (tensor_load_to_lds + s_wait_tensorcnt), shared by all 8 waves per block.
// - A operands stream via per-lane pointers (merged b64 loads), no masks in
//   the K%4==0 hot path (OOB rows/cols only feed unstored C elements).
// ---------------------------------------------------------------------------

typedef __attribute__((ext_vector_type(2))) float v2f;
typedef __attribute__((ext_vector_type(8))) float v8f;
typedef __attribute__((ext_vector_type(4))) unsigned int u32x4;
typedef __attribute__((ext_vector_type(8))) int i32x8;
typedef __attribute__((ext_vector_type(4))) int i32x4;

static __device__ __forceinline__ int imin(int a, int b) { return a < b ? a : b; }

static __device__ __forceinline__ v8f wmma4(v2f a, v2f b, v8f c) {
    return __builtin_amdgcn_wmma_f32_16x16x4_f32(false, a, false, b, (short)0, c,
                                                 false, false);
}

// ---------------------------------------------------------------------------
// TDM: DMA a K x N row-major f32 matrix from global memory into LDS.
// D# built per CDNA5 ISA 8.3/8.4: group0 = {count|flags, lds_addr,
// global_addr, type=2}; group1 = {mask|data_size, dims, tile dims, strides}.
// Issued by wave 0 only (TDM ignores EXEC; one descriptor per instruction).
// Caller must __syncthreads() after (wave 0 waits TENSORcnt==0 first).
// This toolchain exposes the 6-arg builtin (extra i32x8 group, then cpol).
// ---------------------------------------------------------------------------
static __device__ __forceinline__ void tdm_load_to_lds(const float* gptr,
                                                       float* lds, int K, int N) {
    if (threadIdx.x < 32) {
        unsigned long long ga = (unsigned long long)(size_t)gptr;
        unsigned lds_off = (unsigned)(size_t)lds;  // low 32b of LDS aperture addr
        unsigned dim0 = (unsigned)N;               // contiguous dimension
        unsigned dim1 = (unsigned)K;               // rows
        unsigned long long s0 = (unsigned long long)N;  // row stride (elements)

        u32x4 g0;
        g0[0] = 1u;                                         // count=1, user desc
        g0[1] = lds_off;                                    // lds_addr
        g0[2] = (unsigned)ga;                               // global_addr[31:0]
        g0[3] = (unsigned)((ga >> 32) & 0x01FFFFFFu) | (2u << 30); // [56:32]|type=2

        i32x8 g1;
        g1[0] = (int)(2u << 16);                            // data_size=4B
        g1[1] = (int)((dim0 & 0xFFFFu) << 16);              // tensor_dim0[15:0]
        g1[2] = (int)((dim0 >> 16) | ((dim1 & 0xFFFFu) << 16)); // dim0 hi|dim1 lo
        g1[3] = (int)((dim1 >> 16) | ((dim0 & 0xFFFFu) << 16)); // dim1 hi|tile0=N
        g1[4] = (int)(dim1 & 0xFFFFu);                      // tile1=K, tile2=0
        g1[5] = (int)(unsigned)s0;                          // dim0_stride lo32
        g1[6] = (int)(unsigned)(s0 >> 32);                  // stride hi | d1s lo=0
        g1[7] = 0;                                          // dim1_stride hi

        i32x4 z4 = {0, 0, 0, 0};
        i32x8 z8 = {0, 0, 0, 0, 0, 0, 0, 0};
        __builtin_amdgcn_tensor_load_to_lds(g0, g1, z4, z4, z8, 0);
        __builtin_amdgcn_s_wait_tensorcnt(0);
    }
}

// ---------------------------------------------------------------------------
// Column sums of per-batch n x n matrix A:
//   dm[b,j]   = rsqrt(1 + sum_i A[b,i,j])   (DAD normalization, A+I)
//   dinv[b,j] = 1 / sum_i A[b,i,j]          (graph_pool manhattan score)
// ---------------------------------------------------------------------------
__global__ void k_colsum(const float* __restrict__ A, float* __restrict__ dm,
                         float* __restrict__ dinv, int n, int batch) {
    int t = blockIdx.x * blockDim.x + threadIdx.x;
    if (t >= batch * n) return;
    int b = t / n, j = t % n;
    const float* Ab = A + (long long)b * n * n;
    float s = 0.0f;
    for (int i = 0; i < n; ++i) s += Ab[(long long)i * n + j];
    dm[t]   = rsqrtf(s + 1.0f);
    dinv[t] = 1.0f / s;
}

// ---------------------------------------------------------------------------
// Batched WMMA GEMM: C[b] = (relu?) A[b] @ B[b].
// A: MxK (ld=K) streamed from global; B: KxN (ld=N, K*N<=5000) staged in LDS
// via TDM; C: MxN. Batch strides in elements (0 => shared operand).
// 8 waves/block, one 16x16 C tile per wave.
// ---------------------------------------------------------------------------
__global__ void k_gemm(const float* __restrict__ A, const float* __restrict__ B,
                       float* __restrict__ C, int M, int K, int N,
                       long long sA, long long sB, long long sC, int relu) {
    __shared__ float sBuf[5000];
    int b = blockIdx.y;
    const float* Ab = A + (long long)b * sA;
    float*       Cb = C + (long long)b * sC;

    tdm_load_to_lds(B + (long long)b * sB, sBuf, K, N);
    __syncthreads();

    int tN   = (N + 15) >> 4;
    int tM   = (M + 15) >> 4;
    int wave = threadIdx.x >> 5;
    int tile = blockIdx.x * (blockDim.x >> 5) + wave;
    if (tile >= tM * tN) return;
    int mt = tile / tN, nt = tile % tN;

    int lane = threadIdx.x & 31;
    int hi = lane >> 4, lo = lane & 15;
    int row = mt * 16 + lo;
    int col = nt * 16 + lo;
    int rA = imin(row, M - 1);  // clamped: OOB rows feed only unstored C rows
    int cB = imin(col, N - 1);  // clamped: OOB cols feed only unstored C cols

    v8f acc = {};
    if ((K & 3) == 0) {
        // hot path (K = 200 / 100 / 20): mask-free, streaming loads
        const float* ap = Ab + (long long)rA * K + 2 * hi;
        int boff = 2 * hi * N + cB;
        for (int k0 = 0; k0 < K; k0 += 4) {
            v2f af, bf;
            af[0] = ap[0];
            af[1] = ap[1];
            bf[0] = sBuf[boff];
            bf[1] = sBuf[boff + N];
            ap += 4;
            boff += 4 * N;
            acc = wmma4(af, bf, acc);
        }
    } else {
        // generic path (K=25): K-dim masked multiplicatively, loads in-bounds
        for (int k0 = 0; k0 < K; k0 += 4) {
            v2f af, bf;
#pragma unroll
            for (int v = 0; v < 2; ++v) {
                int k  = k0 + v + 2 * hi;
                int kc = imin(k, K - 1);
                float km = (k < K) ? 1.0f : 0.0f;
                af[v] = Ab[(long long)rA * K + kc] * km;
                bf[v] = sBuf[kc * N + cB] * km;
            }
            acc = wmma4(af, bf, acc);
        }
    }
#pragma unroll
    for (int v = 0; v < 8; ++v) {
        int r = mt * 16 + v + 8 * hi;
        int c = nt * 16 + lo;
        if (r < M && c < N) {
            float x = acc[v];
            if (relu) x = fmaxf(x, 0.0f);
            Cb[(long long)r * N + c] = x;
        }
    }
}

// ---------------------------------------------------------------------------
// C[b] = relu( DAD[b] @ B[b] ), DAD[i,k] = dm[i]*(A[i,k]+delta_ik)*dm[k],
// A: n x n per batch (n % 4 == 0). dm and B staged in LDS.
// ---------------------------------------------------------------------------
__global__ void k_gemm_dad(const float* __restrict__ Amat,
                           const float* __restrict__ dm,
                           const float* __restrict__ B, float* __restrict__ C,
                           int n, int N, long long sA, long long sB,
                           long long sC) {
    __shared__ float sBuf[5000];
    __shared__ float sD[256];
    int b = blockIdx.y;
    const float* Ab = Amat + (long long)b * sA;
    float*       Cb = C + (long long)b * sC;

    tdm_load_to_lds(B + (long long)b * sB, sBuf, n, N);
    for (int i = threadIdx.x; i < n; i += (int)blockDim.x)
        sD[i] = dm[(long long)b * n + i];
    __syncthreads();

    int tN   = (N + 15) >> 4;
    int tM   = (n + 15) >> 4;
    int wave = threadIdx.x >> 5;
    int tile = blockIdx.x * (blockDim.x >> 5) + wave;
    if (tile >= tM * tN) return;
    int mt = tile / tN, nt = tile % tN;

    int lane = threadIdx.x & 31;
    int hi = lane >> 4, lo = lane & 15;
    int row = mt * 16 + lo;
    int col = nt * 16 + lo;
    int rA = imin(row, n - 1);
    int cB = imin(col, N - 1);
    float drow = sD[rA];

    const float* ap = Ab + (long long)rA * n + 2 * hi;
    int boff = 2 * hi * N + cB;
    int kk0  = 2 * hi;

    v8f acc = {};
    for (int k0 = 0; k0 < n; k0 += 4) {
        v2f af, bf;
#pragma unroll
        for (int v = 0; v < 2; ++v) {
            int kk = k0 + kk0 + v;
            float a = ap[v] + ((rA == kk) ? 1.0f : 0.0f);  // A + I
            af[v] = a * (drow * sD[kk]);                   // D^-1/2 (A+I) D^-1/2
            bf[v] = sBuf[boff + v * N];
        }
        ap += 4;
        boff += 4 * N;
        acc = wmma4(af, bf, acc);
    }
#pragma unroll
    for (int v = 0; v < 8; ++v) {
        int r = mt * 16 + v + 8 * hi;
        int c = nt * 16 + lo;
        if (r < n && c < N) Cb[(long long)r * N + c] = fmaxf(acc[v], 0.0f);
    }
}

// ---------------------------------------------------------------------------
// graph_pool score: p[b,i] = sum_f | H[b,i,f] - dinv[b,i] * (A@H)[b,i,f] |
// ---------------------------------------------------------------------------
__global__ void k_score(const float* __restrict__ H, const float* __restrict__ AH,
                        const float* __restrict__ dinv, float* __restrict__ p,
                        int n, int F, int batch) {
    int t = blockIdx.x * blockDim.x + threadIdx.x;
    if (t >= batch * n) return;
    float di = dinv[t];
    const float* h  = H  + (long long)t * F;
    const float* ah = AH + (long long)t * F;
    float s = 0.0f;
    for (int f = 0; f < F; ++f) s += fabsf(h[f] - di * ah[f]);
    p[t] = s;
}

// ---------------------------------------------------------------------------
// Stable ascending argsort, keep first topk (rank-count formulation).
// One 256-thread block per batch row, n <= 256.
// ---------------------------------------------------------------------------
__global__ void k_select(const float* __restrict__ p, int* __restrict__ idx,
                         int n, int topk) {
    __shared__ float sp[256];
    int b = blockIdx.x;
    int i = threadIdx.x;
    if (i < n) sp[i] = p[(long long)b * n + i];
    __syncthreads();
    if (i < n) {
        float pi = sp[i];
        int rank = 0;
        for (int j = 0; j < n; ++j) {
            float pj = sp[j];
            rank += (pj < pi) || (pj == pi && j < i);
        }
        if (rank < topk) idx[(long long)b * topk + rank] = i;
    }
}

__global__ void k_gatherH(const float* __restrict__ H, const int* __restrict__ idx,
                          float* __restrict__ out, int n, int F, int topk,
                          int batch) {
    long long t = (long long)blockIdx.x * blockDim.x + threadIdx.x;
    long long total = (long long)batch * topk * F;
    if (t >= total) return;
    int f = (int)(t % F);
    long long r = t / F;
    int rr = (int)(r % topk);
    int b  = (int)(r / topk);
    int src = idx[(long long)b * topk + rr];
    out[t] = H[((long long)b * n + src) * F + f];
}

__global__ void k_gatherA(const float* __restrict__ A, const int* __restrict__ idx,
                          float* __restrict__ out, int n, int topk, int batch) {
    long long t = (long long)blockIdx.x * blockDim.x + threadIdx.x;
    long long total = (long long)batch * topk * topk;
    if (t >= total) return;
    int c = (int)(t % topk);
    long long r = t / topk;
    int rr = (int)(r % topk);
    int b  = (int)(r / topk);
    int sr = idx[(long long)b * topk + rr];
    int sc = idx[(long long)b * topk + c];
    out[t] = A[((long long)b * n + sr) * n + sc];
}

// ---------------------------------------------------------------------------
// Head: gather 10x10 -> flat(100) ++ gender(2) ++ ins(18) ++ age(1),
// feats(121) @ Wd(121x3) + bd, softmax. One thread per batch element.
// ---------------------------------------------------------------------------
__global__ void k_head(const float* __restrict__ H4, const int* __restrict__ idx2,
                       const float* __restrict__ gender,
                       const float* __restrict__ ins,
                       const float* __restrict__ age,
                       const float* __restrict__ Wd, const float* __restrict__ bd,
                       float* __restrict__ out, int batch) {
    int b = blockIdx.x * blockDim.x + threadIdx.x;
    if (b >= batch) return;
    float feats[121];
    for (int r = 0; r < 10; ++r) {
        int src = idx2[b * 10 + r];
        const float* hrow = H4 + ((long long)b * 100 + src) * 10;
        for (int f = 0; f < 10; ++f) feats[r * 10 + f] = hrow[f];
    }
    feats[100] = gender[b * 2 + 0];
    feats[101] = gender[b * 2 + 1];
    for (int k = 0; k < 18; ++k) feats[102 + k] = ins[b * 18 + k];
    feats[120] = age[b];

    float l[3];
    for (int c = 0; c < 3; ++c) {
        float s = bd[c];
        for (int j = 0; j < 121; ++j) s += feats[j] * Wd[j * 3 + c];
        l[c] = s;
    }
    float m = fmaxf(l[0], fmaxf(l[1], l[2]));
    float e0 = __expf(l[0] - m), e1 = __expf(l[1] - m), e2 = __expf(l[2] - m);
    float inv = 1.0f / (e0 + e1 + e2);
    out[b * 3 + 0] = e0 * inv;
    out[b * 3 + 1] = e1 * inv;
    out[b * 3 + 2] = e2 * inv;
}

// ---------------------------------------------------------------------------
extern "C" void kernel_launch(void* const* d_in, const int* in_sizes, int n_in,
                              void* d_out, int out_size, void* d_ws,
                              size_t ws_size, hipStream_t stream) {
    (void)in_sizes; (void)n_in; (void)out_size; (void)ws_size;
    const float* A      = (const float*)d_in[0];  // (1024,200,200)
    const float* X      = (const float*)d_in[1];  // (1024,200,200)
    const float* gender = (const float*)d_in[2];  // (1024,2)
    const float* ins    = (const float*)d_in[3];  // (1024,18)
    const float* age    = (const float*)d_in[4];  // (1024,)
    const float* W0     = (const float*)d_in[5];  // (200,25)
    const float* W1     = (const float*)d_in[6];  // (25,20)
    const float* W2     = (const float*)d_in[7];  // (20,10)
    const float* Wd     = (const float*)d_in[8];  // (121,3)
    const float* bd     = (const float*)d_in[9];  // (3,)
    float* out = (float*)d_out;                   // (1024,3)

    const int NB = 1024, N1 = 200, T0 = 100, T1 = 10;
    const int F0 = 25, F1 = 20, F2 = 10;

    // ---- workspace layout (floats), regions aliased across pipeline phases
    float* ws = (float*)d_ws;
    size_t o = 0;
    float* dm    = ws + o; o += (size_t)NB * N1;
    float* dinv  = ws + o; o += (size_t)NB * N1;
    float* dm3   = ws + o; o += (size_t)NB * T0;
    float* dinv3 = ws + o; o += (size_t)NB * T0;
    float* p1    = ws + o; o += (size_t)NB * N1;
    float* p2    = ws + o; o += (size_t)NB * T0;
    int*   idx1  = (int*)(ws + o); o += (size_t)NB * T0;
    int*   idx2  = (int*)(ws + o); o += (size_t)NB * T1;
    float* regA  = ws + o; o += (size_t)NB * N1 * (F0 + F0); // 10.24M floats
    float* regB  = ws + o; o += (size_t)NB * N1 * F1;        // 4.096M
    float* regC  = ws + o; o += (size_t)NB * N1 * F1;        // 4.096M

    float* XW0 = regA;                         // (1024,200,25)
    float* H1  = regA + (size_t)NB * N1 * F0;  // (1024,200,25)
    float* A3  = regA;                         // (1024,100,100) after H1/XW0 dead
    float* T1b = regB;                         // H1@W1, then A@H2, then H3
    float* H3  = regB;                         // (1024,100,20)
    float* H2  = regC;                         // (1024,200,20)
    float* T3  = regC;                         // H3@W2 after H2 dead
    float* H4  = regC + (size_t)NB * T0 * F2;  // (1024,100,10)
    float* T4  = regC + (size_t)2 * NB * T0 * F2; // A3@H4

    dim3 blk(256);

    // 1) degree vectors from A (n=200)
    k_colsum<<<dim3((NB * N1 + 255) / 256), blk, 0, stream>>>(A, dm, dinv, N1, NB);
    // 2) XW0 = X @ W0          (M=200,K=200,N=25)
    k_gemm<<<dim3(4, NB), blk, 0, stream>>>(X, W0, XW0, N1, N1, F0,
                                            (long long)N1 * N1, 0,
                                            (long long)N1 * F0, 0);
    // 3) H1 = relu(DAD @ XW0)
    k_gemm_dad<<<dim3(4, NB), blk, 0, stream>>>(A, dm, XW0, H1, N1, F0,
                                                (long long)N1 * N1,
                                                (long long)N1 * F0,
                                                (long long)N1 * F0);
    // 4) T1 = H1 @ W1          (M=200,K=25,N=20)
    k_gemm<<<dim3(4, NB), blk, 0, stream>>>(H1, W1, T1b, N1, F0, F1,
                                            (long long)N1 * F0, 0,
                                            (long long)N1 * F1, 0);
    // 5) H2 = relu(DAD @ T1)
    k_gemm_dad<<<dim3(4, NB), blk, 0, stream>>>(A, dm, T1b, H2, N1, F1,
                                                (long long)N1 * N1,
                                                (long long)N1 * F1,
                                                (long long)N1 * F1);
    // 6) AH2 = A @ H2 (into T1b)  (M=200,K=200,N=20)
    k_gemm<<<dim3(4, NB), blk, 0, stream>>>(A, H2, T1b, N1, N1, F1,
                                            (long long)N1 * N1,
                                            (long long)N1 * F1,
                                            (long long)N1 * F1, 0);
    // 7) pool-1 scores + stable top-100 select
    k_score<<<dim3((NB * N1 + 255) / 256), blk, 0, stream>>>(H2, T1b, dinv, p1,
                                                             N1, F1, NB);
    k_select<<<dim3(NB), blk, 0, stream>>>(p1, idx1, N1, T0);
    // 8) gathers: H3 = H2[idx], A3 = A[idx][:,idx]
    k_gatherH<<<dim3((NB * T0 * F1 + 255) / 256), blk, 0, stream>>>(
        H2, idx1, H3, N1, F1, T0, NB);
    k_gatherA<<<dim3((NB * T0 * T0 + 255) / 256), blk, 0, stream>>>(
        A, idx1, A3, N1, T0, NB);
    // 9) degree vectors from A3 (n=100)
    k_colsum<<<dim3((NB * T0 + 255) / 256), blk, 0, stream>>>(A3, dm3, dinv3,
                                                              T0, NB);
    // 10) T3 = H3 @ W2         (M=100,K=20,N=10)
    k_gemm<<<dim3(1, NB), blk, 0, stream>>>(H3, W2, T3, T0, F1, F2,
                                            (long long)T0 * F1, 0,
                                            (long long)T0 * F2, 0);
    // 11) H4 = relu(DAD3 @ T3)
    k_gemm_dad<<<dim3(1, NB), blk, 0, stream>>>(A3, dm3, T3, H4, T0, F2,
                                                (long long)T0 * T0,
                                                (long long)T0 * F2,
                                                (long long)T0 * F2);
    // 12) AH4 = A3 @ H4 (into T4)
    k_gemm<<<dim3(1, NB), blk, 0, stream>>>(A3, H4, T4, T0, T0, F2,
                                            (long long)T0 * T0,
                                            (long long)T0 * F2,
                                            (long long)T0 * F2, 0);
    // 13) pool-2 scores + top-10 select
    k_score<<<dim3((NB * T0 + 255) / 256), blk, 0, stream>>>(H4, T4, dinv3, p2,
                                                             T0, F2, NB);
    k_select<<<dim3(NB), blk, 0, stream>>>(p2, idx2, T0, T1);
    // 14) head
    k_head<<<dim3((NB + 255) / 256), blk, 0, stream>>>(H4, idx2, gender, ins,
                                                       age, Wd, bd, out, NB);
}